// AutoEncoder_38465727102983
// MI455X (gfx1250) — compile-verified
//
#include <hip/hip_runtime.h>

typedef __attribute__((ext_vector_type(2))) float v2f;
typedef __attribute__((ext_vector_type(8))) float v8f;

#define FEAT_MAX 88   // max layer width
#define HPAD     96   // padded row stride for all workspace feature buffers

// ---------------- degree / norm ----------------
__global__ __launch_bounds__(256) void k_fill1(float* __restrict__ p, int n) {
  int i = blockIdx.x * 256 + threadIdx.x;
  if (i < n) p[i] = 1.0f;  // self-loop contributes 1 to every node's degree
}

__global__ __launch_bounds__(256) void k_degcount(const int* __restrict__ dst,
                                                  float* __restrict__ deg, int nE) {
  int e = blockIdx.x * 256 + threadIdx.x;
  if (e < nE) atomicAdd(&deg[dst[e]], 1.0f);
}

__global__ __launch_bounds__(256) void k_rsqrt(float* __restrict__ p, int n) {
  int i = blockIdx.x * 256 + threadIdx.x;
  if (i < n) p[i] = rsqrtf(p[i]);  // deg >= 1 always
}

// ---------------- GEMM + dinv row-scale epilogue (WMMA f32 16x16x4) ----------------
// G[v,:] = dinv[v] * (H[v,:] @ W);  S = G  (self-loop seed for scatter accumulator)
// H has row stride hstride with zero-padded columns [din, roundup4(din)).
// G,S have row stride HPAD (compact cols [0,dout) valid).
__global__ __launch_bounds__(256)
void k_gemm_scale(const float* __restrict__ H, int hstride,
                  const float* __restrict__ W,
                  const float* __restrict__ dinv, float* __restrict__ G,
                  float* __restrict__ S, int n, int din, int dout) {
  // k-pair packed weight slice: element (kp=k/2, c, j=k&1) at [kp*32 + c*2 + j]
  __shared__ float wtile[FEAT_MAX * 16];

  const int din4 = (din + 3) & ~3;
  const int tid  = threadIdx.x;
  const int col0 = (int)blockIdx.y << 4;

  for (int i = tid; i < din4 * 16; i += 256) {
    const int k = i >> 4, c = i & 15;
    const int col = col0 + c;
    float v = 0.0f;
    if (k < din && col < dout) v = W[(size_t)k * dout + col];
    wtile[((k >> 1) << 5) + (c << 1) + (k & 1)] = v;
  }
  __syncthreads();

  const int wave = tid >> 5;
  const int lane = tid & 31;
  const int half = lane >> 4;   // 0: lanes 0-15, 1: lanes 16-31
  const int idx  = lane & 15;

  const int row0   = (blockIdx.x * 8 + wave) << 4;
  const bool active = (row0 < n);
  const int rowb   = active ? row0 : 0;     // clamped base so loads stay in range
  const int col    = col0 + idx;

  const float* ap = H + (size_t)(rowb + idx) * hstride + (half << 1);
  const float* wp = wtile + (half << 5) + (idx << 1);

  v8f acc = {};
  for (int k0 = 0; k0 < din4; k0 += 4) {
    const v2f a = *(const v2f*)(ap + k0);          // global_load_b64, unconditional
    const v2f b = *(const v2f*)(wp + (k0 << 4));   // ds_load_b64, conflict-free
    acc = __builtin_amdgcn_wmma_f32_16x16x4_f32(
        false, a, false, b, (short)0, acc, false, false);
  }

  if (active && col < dout) {
#pragma unroll
    for (int r = 0; r < 8; ++r) {
      const int row  = row0 + r + (half << 3);     // D: VGPR r -> M = r + 8*half
      const float gv = dinv[row] * acc[r];
      const size_t o = (size_t)row * HPAD + col;
      G[o] = gv;
      S[o] = gv;
    }
  }
}

// ---------------- edge scatter: S[dst,:] += G[src,:] (stride HPAD) ----------------
__global__ __launch_bounds__(256)
void k_scatter(const int* __restrict__ srcI, const int* __restrict__ dstI,
               const float* __restrict__ G, float* __restrict__ S, int nE, int d) {
  int e = blockIdx.x * 256 + threadIdx.x;
  if (e >= nE) return;
  const float* gp = G + (size_t)srcI[e] * HPAD;   // 384B-aligned rows
  float*       sp = S + (size_t)dstI[e] * HPAD;
  const int d4 = d & ~3;
  for (int f = 0; f < d4; f += 4) {
    const float4 gv = *(const float4*)(gp + f);   // global_load_b128
    atomicAdd(sp + f + 0, gv.x);
    atomicAdd(sp + f + 1, gv.y);
    atomicAdd(sp + f + 2, gv.z);
    atomicAdd(sp + f + 3, gv.w);
  }
  for (int f = d4; f < d; ++f) atomicAdd(sp + f, gp[f]);
}

// ---------------- out[v,f] = act(dinv[v]*S[v,f] + b[f]); zero-fill pad cols ----------
__global__ __launch_bounds__(256)
void k_finalize(const float* __restrict__ S, const float* __restrict__ dinv,
                const float* __restrict__ bias, float* __restrict__ Out,
                int n, int d, int dpad, int ostride, int relu) {
  unsigned tid = blockIdx.x * 256u + threadIdx.x;
  const unsigned total = (unsigned)n * (unsigned)dpad;
  if (tid >= total) return;
  const unsigned v = tid / (unsigned)dpad;
  const unsigned f = tid - v * (unsigned)dpad;
  float val = 0.0f;
  if (f < (unsigned)d) {
    val = fmaf(dinv[v], S[(size_t)v * HPAD + f], bias[f]);
    if (relu) val = fmaxf(val, 0.0f);
  }
  Out[(size_t)v * ostride + f] = val;
}

static inline int cdiv(int a, int b) { return (a + b - 1) / b; }

extern "C" void kernel_launch(void* const* d_in, const int* in_sizes, int n_in,
                              void* d_out, int out_size, void* d_ws, size_t ws_size,
                              hipStream_t stream) {
  (void)n_in; (void)out_size; (void)ws_size;

  const float* x  = (const float*)d_in[0];
  const float* Ws[4] = {(const float*)d_in[1], (const float*)d_in[3],
                        (const float*)d_in[5], (const float*)d_in[7]};
  const float* bs[4] = {(const float*)d_in[2], (const float*)d_in[4],
                        (const float*)d_in[6], (const float*)d_in[8]};
  const int* ei = (const int*)d_in[9];

  const int E = in_sizes[9] / 2;
  const int n = in_sizes[0] / 88;
  const int* src = ei;        // edge_index[0]
  const int* dst = ei + E;    // edge_index[1]

  float* dinv = (float*)d_ws;                       // [n]
  float* G    = dinv + n;                           // [n * HPAD]
  float* S    = G + (size_t)n * HPAD;               // [n * HPAD]
  float* H    = S + (size_t)n * HPAD;               // [n * HPAD]

  // normalization: dinv = rsqrt(1 + indegree)
  k_fill1<<<cdiv(n, 256), 256, 0, stream>>>(dinv, n);
  k_degcount<<<cdiv(E, 256), 256, 0, stream>>>(dst, dinv, E);
  k_rsqrt<<<cdiv(n, 256), 256, 0, stream>>>(dinv, n);

  const int dims[5]  = {88, 65, 50, 65, 88};
  const int relus[4] = {1, 0, 1, 0};

  const float* hin = x;
  int hstride = 88;                                 // x is compact; 88 % 4 == 0
  for (int L = 0; L < 4; ++L) {
    const int din = dims[L], dout = dims[L + 1];
    const int dpad = (L == 3) ? 88 : ((dout + 3) & ~3);   // K-pad for next layer
    dim3 gg(cdiv(cdiv(n, 16), 8), cdiv(dout, 16));
    k_gemm_scale<<<gg, 256, 0, stream>>>(hin, hstride, Ws[L], dinv, G, S, n, din, dout);
    k_scatter<<<cdiv(E, 256), 256, 0, stream>>>(src, dst, G, S, E, dout);
    float* out = (L == 3) ? (float*)d_out : H;
    const int ostride = (L == 3) ? 88 : HPAD;
    k_finalize<<<cdiv(n * dpad, 256), 256, 0, stream>>>(S, dinv, bs[L], out,
                                                        n, dout, dpad, ostride, relus[L]);
    hin = H;
    hstride = HPAD;
  }
}